// HypER_12730283066031
// MI455X (gfx1250) — compile-verified
//
#include <hip/hip_runtime.h>

// ---------------- CDNA5 fp32 WMMA helpers ----------------
typedef __attribute__((ext_vector_type(2))) float v2f;
typedef __attribute__((ext_vector_type(8))) float v8f;

__device__ __forceinline__ v8f wmma4(v2f a, v2f b, v8f c) {
  // V_WMMA_F32_16X16X4_F32 : D(16x16 f32) = A(16x4 f32) x B(4x16 f32) + C
  return __builtin_amdgcn_wmma_f32_16x16x4_f32(
      /*neg_a=*/false, a, /*neg_b=*/false, b,
      /*c_mod=*/(short)0, c, /*reuse_a=*/false, /*reuse_b=*/false);
}

// ---------------- problem constants ----------------
#define BSAMP 8192
#define DDIM  400
#define OCH   32
#define FWIN  9
#define WOUT  392           // DDIM - FWIN + 1
#define KLEN  288           // OCH*FWIN
#define FEAT  12544         // OCH*WOUT
#define EPSV  1e-5f

// stats layout inside workspace (floats)
#define ST_BN0SUM 0
#define ST_BN0SQ  1
#define ST_BN1SUM 2
#define ST_BN1SQ  34
#define ST_BN2SUM 66
#define ST_BN2SQ  466
#define ST_TOTAL  1024

// ---------------- generic 32x32 tiled transpose ----------------
// dst[c*rows + r] = src[r*cols + c]
__global__ __launch_bounds__(256) void hyper_transpose_kernel(
    const float* __restrict__ src, float* __restrict__ dst, int rows, int cols) {
  __shared__ float tile[32][33];
  int c0 = blockIdx.x * 32, r0 = blockIdx.y * 32;
  int tx = threadIdx.x & 31, ty = threadIdx.x >> 5;   // 8 rows of 32 lanes
  for (int i = ty; i < 32; i += 8) {
    int r = r0 + i, c = c0 + tx;
    tile[i][tx] = (r < rows && c < cols) ? src[r * cols + c] : 0.f;
  }
  __syncthreads();
  for (int i = ty; i < 32; i += 8) {
    int c = c0 + i, r = r0 + tx;
    if (c < cols && r < rows) dst[c * rows + r] = tile[tx][i];
  }
}

// ---------------- gathered GEMM: HT[r] = uee[ids[r]] @ W_E ----------------
// M = 16384 (1024 tiles), N = 400 (25 tiles), K = 400. One 16x16 tile / wave.
__global__ __launch_bounds__(256) void hyper_gather_gemm_kernel(
    const int* __restrict__ ids, const float* __restrict__ uee,
    const float* __restrict__ WE, float* __restrict__ HT) {
  int wvg   = blockIdx.x * 8 + (threadIdx.x >> 5);
  int tileM = wvg / 25;
  int tileN = wvg % 25;
  int lane = threadIdx.x & 31, lm = lane & 15, hs = lane >> 4;

  long src = (long)ids[tileM * 16 + lm] * DDIM;   // gathered row base
  int  n   = tileN * 16 + lm;

  v8f c = {0.f,0.f,0.f,0.f,0.f,0.f,0.f,0.f};
#pragma unroll 4
  for (int k0 = 0; k0 < DDIM; k0 += 4) {
    int kk = k0 + hs * 2;
    v2f a = *(const v2f*)(uee + src + kk);            // A[M=lm][K=kk,kk+1]
    v2f b; b.x = WE[kk * DDIM + n];                   // B[K][N], coalesced in n
           b.y = WE[(kk + 1) * DDIM + n];
    c = wmma4(a, b, c);
  }
  int rbase = tileM * 16 + hs * 8;
#pragma unroll
  for (int r = 0; r < 8; r++) HT[(rbase + r) * DDIM + n] = c[r];
}

// ---------------- BN0 scalar stats over head (first 8192 rows of HT) -------
__global__ __launch_bounds__(256) void hyper_bn0_reduce_kernel(
    const float* __restrict__ HT, float* __restrict__ stats) {
  __shared__ float rs[256], rq[256];
  float s = 0.f, q = 0.f;
  const int total = BSAMP * DDIM;
  for (int i = blockIdx.x * 256 + threadIdx.x; i < total; i += gridDim.x * 256) {
    float v = HT[i]; s += v; q += v * v;
  }
  rs[threadIdx.x] = s; rq[threadIdx.x] = q;
  __syncthreads();
  for (int off = 128; off > 0; off >>= 1) {
    if ((int)threadIdx.x < off) {
      rs[threadIdx.x] += rs[threadIdx.x + off];
      rq[threadIdx.x] += rq[threadIdx.x + off];
    }
    __syncthreads();
  }
  if (threadIdx.x == 0) {
    atomicAdd(&stats[ST_BN0SUM], rs[0]);
    atomicAdd(&stats[ST_BN0SQ],  rq[0]);
  }
}

// ---------------- KM = tail @ fc1_w^T + fc1_b  (8192 x 288) ---------------
// M = 8192 (512 tiles), N = 288 (18 tiles), K = 400. B = FC1WT (400x288).
__global__ __launch_bounds__(256) void hyper_kmat_gemm_kernel(
    const float* __restrict__ HT, const float* __restrict__ FC1WT,
    const float* __restrict__ fc1_b, float* __restrict__ KM) {
  int wvg   = blockIdx.x * 8 + (threadIdx.x >> 5);
  int tileM = wvg / 18;
  int tileN = wvg % 18;
  int lane = threadIdx.x & 31, lm = lane & 15, hs = lane >> 4;

  long arow = (long)(BSAMP + tileM * 16 + lm) * DDIM;   // tail rows
  int  n    = tileN * 16 + lm;

  v8f c = {0.f,0.f,0.f,0.f,0.f,0.f,0.f,0.f};
#pragma unroll 4
  for (int k0 = 0; k0 < DDIM; k0 += 4) {
    int kk = k0 + hs * 2;
    v2f a = *(const v2f*)(HT + arow + kk);
    v2f b; b.x = FC1WT[kk * KLEN + n];
           b.y = FC1WT[(kk + 1) * KLEN + n];
    c = wmma4(a, b, c);
  }
  float bias = fc1_b[n];
  int rbase = tileM * 16 + hs * 8;
#pragma unroll
  for (int r = 0; r < 8; r++) KM[(rbase + r) * KLEN + n] = c[r] + bias;
}

// ---------------- BN1 per-channel stats of conv (conv not stored) ----------
__global__ __launch_bounds__(256) void hyper_bn1_stats_kernel(
    const float* __restrict__ HT, const float* __restrict__ KM,
    const float* __restrict__ bn0_g, const float* __restrict__ bn0_b,
    float* __restrict__ stats) {
  __shared__ float xr[DDIM];
  __shared__ float kr[KLEN];
  int b = blockIdx.x, tid = threadIdx.x;

  const float cnt0  = (float)(BSAMP * DDIM);
  float mean0 = stats[ST_BN0SUM] / cnt0;
  float var0  = stats[ST_BN0SQ] / cnt0 - mean0 * mean0;
  float a0 = rsqrtf(var0 + EPSV) * bn0_g[0];
  float c0 = bn0_b[0] - mean0 * a0;

  for (int d = tid; d < DDIM; d += 256) xr[d] = HT[b * DDIM + d] * a0 + c0;
  for (int d = tid; d < KLEN; d += 256) kr[d] = KM[b * KLEN + d];
  __syncthreads();

  int o = tid >> 3, sub = tid & 7;        // 8 threads per output channel
  float k9[FWIN];
#pragma unroll
  for (int w = 0; w < FWIN; w++) k9[w] = kr[o * FWIN + w];
  float s = 0.f, q = 0.f;
  for (int j = sub; j < WOUT; j += 8) {
    float acc = 0.f;
#pragma unroll
    for (int w = 0; w < FWIN; w++) acc += xr[j + w] * k9[w];
    s += acc; q += acc * acc;
  }
  s += __shfl_down(s, 4, 8); s += __shfl_down(s, 2, 8); s += __shfl_down(s, 1, 8);
  q += __shfl_down(q, 4, 8); q += __shfl_down(q, 2, 8); q += __shfl_down(q, 1, 8);
  if (sub == 0) {
    atomicAdd(&stats[ST_BN1SUM + o], s);
    atomicAdd(&stats[ST_BN1SQ  + o], q);
  }
}

// ---------------- big GEMM: out_raw = bn1(conv) @ fc_w^T + fc_b ------------
// One block per 16 output rows; conv strips recomputed in LDS per channel.
// M-tile = 16, N = 400 (25 tiles over 8 waves), K = 12544 done as 32 x 392.
__global__ __launch_bounds__(256) void hyper_fc_gemm_kernel(
    const float* __restrict__ HT, const float* __restrict__ KM,
    const float* __restrict__ FCWT, const float* __restrict__ fc_b,
    const float* __restrict__ bn0_g, const float* __restrict__ bn0_b,
    const float* __restrict__ bn1_g, const float* __restrict__ bn1_b,
    const float* __restrict__ stats, float* __restrict__ OUTRAW) {
  __shared__ float xr[16 * DDIM];         // BN0-normalized x rows
  __shared__ float convo[16 * 402];       // normalized conv strip, stride 402

  int tid = threadIdx.x;
  int m0row = blockIdx.x * 16;

  const float cnt0 = (float)(BSAMP * DDIM);
  float mean0 = stats[ST_BN0SUM] / cnt0;
  float var0  = stats[ST_BN0SQ] / cnt0 - mean0 * mean0;
  float a0 = rsqrtf(var0 + EPSV) * bn0_g[0];
  float c0 = bn0_b[0] - mean0 * a0;

  for (int idx = tid; idx < 16 * DDIM; idx += 256) {
    int r = idx / DDIM, d = idx - r * DDIM;
    xr[idx] = HT[(m0row + r) * DDIM + d] * a0 + c0;
  }

  int lane = tid & 31, wv = tid >> 5, lm = lane & 15, hs = lane >> 4;
  int rr = tid >> 4, jlane = tid & 15;    // compute-phase mapping: 16 thr/row

  v8f z = {0.f,0.f,0.f,0.f,0.f,0.f,0.f,0.f};
  v8f acc[4] = {z, z, z, z};

  const float cnt1 = (float)(BSAMP * WOUT);

  for (int o = 0; o < OCH; o++) {
    float m1 = stats[ST_BN1SUM + o] / cnt1;
    float v1 = stats[ST_BN1SQ  + o] / cnt1 - m1 * m1;
    float a1 = rsqrtf(v1 + EPSV) * bn1_g[o];
    float c1 = bn1_b[o] - m1 * a1;
    float k9[FWIN];
#pragma unroll
    for (int w = 0; w < FWIN; w++) k9[w] = KM[(m0row + rr) * KLEN + o * FWIN + w];

    __syncthreads();                       // readers of previous strip done
    for (int j = jlane; j < WOUT; j += 16) {
      float s = 0.f;
#pragma unroll
      for (int w = 0; w < FWIN; w++) s += xr[rr * DDIM + j + w] * k9[w];
      convo[rr * 402 + j] = s * a1 + c1;   // BN1 applied on the fly
    }
    __syncthreads();                       // strip visible to all waves

    for (int j0 = 0; j0 < WOUT; j0 += 4) {
      int jj  = j0 + hs * 2;
      v2f a   = *(const v2f*)&convo[lm * 402 + jj];   // 8B aligned ds_load_b64
      int kkg = o * WOUT + jj;
      if (j0 + 8 < WOUT)
        __builtin_prefetch(&FCWT[(kkg + 8) * DDIM + lm], 0, 0);
#pragma unroll
      for (int i = 0; i < 4; i++) {
        int t = wv + 8 * i;                // wave-uniform guard: EXEC stays full
        if (t < 25) {
          int n = t * 16 + lm;
          v2f b; b.x = FCWT[kkg * DDIM + n];
                 b.y = FCWT[(kkg + 1) * DDIM + n];
          acc[i] = wmma4(a, b, acc[i]);
        }
      }
    }
  }

  // bias, store raw output, accumulate BN2 column stats
#pragma unroll
  for (int i = 0; i < 4; i++) {
    int t = wv + 8 * i;
    if (t < 25) {
      int n = t * 16 + lm;
      float bias = fc_b[n];
      float s = 0.f, q = 0.f;
#pragma unroll
      for (int r = 0; r < 8; r++) {
        int gr = m0row + r + hs * 8;
        float val = acc[i][r] + bias;
        OUTRAW[gr * DDIM + n] = val;
        s += val; q += val * val;
      }
      atomicAdd((float*)&stats[ST_BN2SUM + n], s);
      atomicAdd((float*)&stats[ST_BN2SQ  + n], q);
    }
  }
}

// ---------------- BN2 + ReLU -> final output ----------------
__global__ __launch_bounds__(256) void hyper_bn2_apply_kernel(
    const float* __restrict__ OUTRAW, const float* __restrict__ stats,
    const float* __restrict__ g2, const float* __restrict__ b2,
    float* __restrict__ out) {
  int idx = blockIdx.x * 256 + threadIdx.x;
  if (idx >= BSAMP * DDIM) return;
  int c = idx % DDIM;
  float m   = stats[ST_BN2SUM + c] / (float)BSAMP;
  float var = stats[ST_BN2SQ  + c] / (float)BSAMP - m * m;
  float a = rsqrtf(var + EPSV) * g2[c];
  float v = (OUTRAW[idx] - m) * a + b2[c];
  out[idx] = fmaxf(v, 0.f);
}

// ---------------- host orchestration ----------------
extern "C" void kernel_launch(void* const* d_in, const int* in_sizes, int n_in,
                              void* d_out, int out_size, void* d_ws, size_t ws_size,
                              hipStream_t stream) {
  const int*   ids   = (const int*)d_in[0];
  const float* uee   = (const float*)d_in[1];
  const float* WE    = (const float*)d_in[2];
  const float* fc1_w = (const float*)d_in[3];
  const float* fc1_b = (const float*)d_in[4];
  const float* fc_w  = (const float*)d_in[5];
  const float* fc_b  = (const float*)d_in[6];
  const float* bn0_g = (const float*)d_in[7];
  const float* bn0_b = (const float*)d_in[8];
  const float* bn1_g = (const float*)d_in[9];
  const float* bn1_b = (const float*)d_in[10];
  const float* bn2_g = (const float*)d_in[11];
  const float* bn2_b = (const float*)d_in[12];
  float* out = (float*)d_out;

  float* ws     = (float*)d_ws;
  float* stats  = ws;                                   // 1024 f
  float* HT     = ws + ST_TOTAL;                        // 16384*400
  float* KM     = HT + (size_t)16384 * DDIM;            // 8192*288
  float* FCWT   = KM + (size_t)BSAMP * KLEN;            // 12544*400
  float* FC1WT  = FCWT + (size_t)FEAT * DDIM;           // 400*288
  float* OUTRAW = FC1WT + (size_t)DDIM * KLEN;          // 8192*400

  hipMemsetAsync(stats, 0, ST_TOTAL * sizeof(float), stream);

  dim3 blk(256);
  hyper_transpose_kernel<<<dim3((FEAT + 31) / 32, (DDIM + 31) / 32), blk, 0, stream>>>(
      fc_w, FCWT, DDIM, FEAT);
  hyper_transpose_kernel<<<dim3((DDIM + 31) / 32, (KLEN + 31) / 32), blk, 0, stream>>>(
      fc1_w, FC1WT, KLEN, DDIM);

  hyper_gather_gemm_kernel<<<(1024 * 25) / 8, blk, 0, stream>>>(ids, uee, WE, HT);
  hyper_bn0_reduce_kernel<<<512, blk, 0, stream>>>(HT, stats);
  hyper_kmat_gemm_kernel<<<(512 * 18) / 8, blk, 0, stream>>>(HT, FC1WT, fc1_b, KM);
  hyper_bn1_stats_kernel<<<BSAMP, blk, 0, stream>>>(HT, KM, bn0_g, bn0_b, stats);
  hyper_fc_gemm_kernel<<<BSAMP / 16, blk, 0, stream>>>(
      HT, KM, FCWT, fc_b, bn0_g, bn0_b, bn1_g, bn1_b, stats, OUTRAW);
  hyper_bn2_apply_kernel<<<(BSAMP * DDIM + 255) / 256, blk, 0, stream>>>(
      OUTRAW, stats, bn2_g, bn2_b, out);
}